// OpenPairIndexer_34514357190720
// MI455X (gfx1250) — compile-verified
//
#include <hip/hip_runtime.h>
#include <math.h>

// ---------------------------------------------------------------------------
// OpenPairIndexer for MI455X (gfx1250, wave32)
//
// Perf model: output is 6*MAX_PAIRS*4B = 403 MB, ~97% zero padding -> the
// kernel is store-bandwidth bound (~17us @ 23.3 TB/s). Inputs (393KB) are L2
// resident. Distance Gram matrices are computed with V_WMMA_F32_16X16X4_F32
// (K=4 = xyz + zero pad), chosen over f16/bf16 WMMA because |r|^2 ~ O(1e3)
// would lose the cutoff boundary in 8-11 bit mantissas.
//
// Output layout (4-byte elements, JAX x64-disabled => int32 indices):
//   [ dist : P f32 ][ pair_first : P i32 ][ pair_second : P i32 ][ paircoord : 3P f32 ]
// ---------------------------------------------------------------------------

#define N_MOL    64
#define N_ATOMS  512
#define NROW     (N_MOL * N_ATOMS)                       // 32768 rows (m,i)
#define MAX_PAIRS ((size_t)N_MOL * N_ATOMS * N_ATOMS)    // 16777216
#define CUTOFF2  36.0f
#define BLANK_BIG 1.0e30f

typedef __attribute__((ext_vector_type(2))) float v2f;
typedef __attribute__((ext_vector_type(8))) float v8f;

// ---------------------------------------------------------------------------
// Kernel 1: zero-fill d_out (padding of the nonzero-compaction must be 0).
// b128 stores, grid-stride; this is the bandwidth-critical kernel.
// ---------------------------------------------------------------------------
__global__ void zero_out(float4* p4, long long n4, float* p, long long n)
{
    long long i = (long long)blockIdx.x * blockDim.x + threadIdx.x;
    const long long stride = (long long)gridDim.x * blockDim.x;
    const float4 z = make_float4(0.f, 0.f, 0.f, 0.f);
    for (; i < n4; i += stride) p4[i] = z;
    if (blockIdx.x == 0 && threadIdx.x == 0)
        for (long long k = n4 * 4; k < n; ++k) p[k] = 0.f;
}

// ---------------------------------------------------------------------------
// Kernel 2: per-row neighbor counts via FP32 WMMA Gram tiles.
// One wave per (mol, itile of 16 rows). A/B fragment layout (ISA 7.12.2,
// 32-bit 16x4 A documented; 4x16 B inferred by symmetry):
//   lanes 0-15 : (x, y) of atom (tile*16 + lane)      [K=0, K=1]
//   lanes 16-31: (z, 0) of atom (tile*16 + lane-16)   [K=2, K=3]
// D layout: VGPR v -> row M = v (lanes 0-15) / v+8 (lanes 16-31), N = lane&15.
// ---------------------------------------------------------------------------
__global__ __launch_bounds__(32) void pair_count_wmma(
    const float* __restrict__ coords,
    const unsigned char* __restrict__ nonblank,
    int* __restrict__ counts)
{
    const int lane  = threadIdx.x;
    const int hi    = lane >> 4;      // 0: lanes 0-15, 1: lanes 16-31
    const int l15   = lane & 15;
    const int itile = blockIdx.x;     // 0..31
    const int mol   = blockIdx.y;     // 0..63

    const float* cm = coords + (size_t)mol * N_ATOMS * 3;
    const unsigned char* nbm = nonblank + (size_t)mol * N_ATOMS;

    const int irow = itile * 16 + l15;
    const float ix = cm[irow * 3 + 0];
    const float iy = cm[irow * 3 + 1];
    const float iz = cm[irow * 3 + 2];
    float r2i_self = ix * ix + iy * iy + iz * iz;
    if (!nbm[irow]) r2i_self = BLANK_BIG;   // blank atom -> never within cutoff

    v2f afrag;
    afrag.x = hi ? iz : ix;
    afrag.y = hi ? 0.0f : iy;

    // broadcast |r_i|^2 of row (v + 8*hi) from the lane that loaded it
    float r2i[8];
#pragma unroll
    for (int v = 0; v < 8; ++v) r2i[v] = __shfl(r2i_self, v + 8 * hi, 32);

    int cnt[8];
#pragma unroll
    for (int v = 0; v < 8; ++v) cnt[v] = 0;

    for (int jtile = 0; jtile < N_ATOMS / 16; ++jtile) {
        const int jrow = jtile * 16 + l15;
        const float jx = cm[jrow * 3 + 0];
        const float jy = cm[jrow * 3 + 1];
        const float jz = cm[jrow * 3 + 2];
        float r2j = jx * jx + jy * jy + jz * jz;
        if (!nbm[jrow]) r2j = BLANK_BIG;

        v2f bfrag;
        bfrag.x = hi ? jz : jx;
        bfrag.y = hi ? 0.0f : jy;

        v8f c = {0.f, 0.f, 0.f, 0.f, 0.f, 0.f, 0.f, 0.f};
        v8f d = __builtin_amdgcn_wmma_f32_16x16x4_f32(
            false, afrag, false, bfrag, (short)0, c, false, false);

#pragma unroll
        for (int v = 0; v < 8; ++v) {
            const int   im = itile * 16 + v + 8 * hi;
            const float d2 = r2i[v] + r2j - 2.0f * d[v];
            cnt[v] += ((d2 < CUTOFF2) && (im != jrow)) ? 1 : 0;
        }
    }

    // reduce each row-count across its 16-lane half (xor<16 stays in half)
#pragma unroll
    for (int v = 0; v < 8; ++v) {
        int cc = cnt[v];
        cc += __shfl_xor(cc, 1, 32);
        cc += __shfl_xor(cc, 2, 32);
        cc += __shfl_xor(cc, 4, 32);
        cc += __shfl_xor(cc, 8, 32);
        cnt[v] = cc;
    }
    if (l15 == 0) {
        const int base = mol * N_ATOMS + itile * 16 + 8 * hi;
#pragma unroll
        for (int v = 0; v < 8; ++v) counts[base + v] = cnt[v];
    }
}

// ---------------------------------------------------------------------------
// Kernel 3: exclusive prefix sum over 32768 row counts (row-major (m,i) order
// == jnp.nonzero emission order). Single 1024-thread block, two-level scan.
// ---------------------------------------------------------------------------
__global__ __launch_bounds__(1024) void scan_rows(
    const int* __restrict__ counts, int* __restrict__ offsets)
{
    __shared__ int lsum[1024];
    const int t = threadIdx.x;
    const int chunk = NROW / 1024;    // 32
    const int base = t * chunk;

    int s = 0;
    for (int k = 0; k < chunk; ++k) s += counts[base + k];
    lsum[t] = s;
    __syncthreads();

    for (int off = 1; off < 1024; off <<= 1) {
        int v = (t >= off) ? lsum[t - off] : 0;
        __syncthreads();
        lsum[t] += v;
        __syncthreads();
    }

    int run = (t == 0) ? 0 : lsum[t - 1];
    for (int k = 0; k < chunk; ++k) {
        offsets[base + k] = run;
        run += counts[base + k];
    }
}

// ---------------------------------------------------------------------------
// Kernel 4: emit pairs. Identical WMMA tile + mask arithmetic as the count
// kernel (so counts/offsets agree exactly), then ordered within-row
// compaction via 16-lane shfl_up prefix sums. Output geometry is recomputed
// through the real_atoms / inv_real_atoms gathers, as in the reference.
// ---------------------------------------------------------------------------
__global__ __launch_bounds__(32) void pair_emit_wmma(
    const float* __restrict__ coords,
    const unsigned char* __restrict__ nonblank,
    const int* __restrict__ real_atoms,
    const int* __restrict__ inv_real_atoms,
    const int* __restrict__ offsets,
    float* __restrict__ dist,
    int* __restrict__ pfirst,
    int* __restrict__ psecond,
    float* __restrict__ pcoord)
{
    const int lane  = threadIdx.x;
    const int hi    = lane >> 4;
    const int l15   = lane & 15;
    const int itile = blockIdx.x;
    const int mol   = blockIdx.y;

    const float* cm = coords + (size_t)mol * N_ATOMS * 3;
    const unsigned char* nbm = nonblank + (size_t)mol * N_ATOMS;

    const int irow = itile * 16 + l15;
    const float ix = cm[irow * 3 + 0];
    const float iy = cm[irow * 3 + 1];
    const float iz = cm[irow * 3 + 2];
    float r2i_self = ix * ix + iy * iy + iz * iz;
    if (!nbm[irow]) r2i_self = BLANK_BIG;

    v2f afrag;
    afrag.x = hi ? iz : ix;
    afrag.y = hi ? 0.0f : iy;

    float r2i[8];
#pragma unroll
    for (int v = 0; v < 8; ++v) r2i[v] = __shfl(r2i_self, v + 8 * hi, 32);

    int base[8];
#pragma unroll
    for (int v = 0; v < 8; ++v)
        base[v] = offsets[mol * N_ATOMS + itile * 16 + v + 8 * hi];

    for (int jtile = 0; jtile < N_ATOMS / 16; ++jtile) {
        const int jrow = jtile * 16 + l15;
        const float jx = cm[jrow * 3 + 0];
        const float jy = cm[jrow * 3 + 1];
        const float jz = cm[jrow * 3 + 2];
        float r2j = jx * jx + jy * jy + jz * jz;
        if (!nbm[jrow]) r2j = BLANK_BIG;

        v2f bfrag;
        bfrag.x = hi ? jz : jx;
        bfrag.y = hi ? 0.0f : jy;

        v8f c = {0.f, 0.f, 0.f, 0.f, 0.f, 0.f, 0.f, 0.f};
        v8f d = __builtin_amdgcn_wmma_f32_16x16x4_f32(
            false, afrag, false, bfrag, (short)0, c, false, false);

#pragma unroll
        for (int v = 0; v < 8; ++v) {
            const int   im  = itile * 16 + v + 8 * hi;
            const float d2  = r2i[v] + r2j - 2.0f * d[v];
            const bool  hit = (d2 < CUTOFF2) && (im != jrow);

            // inclusive prefix sum of hits within this 16-lane half (j ascending)
            int inc = hit ? 1 : 0;
#pragma unroll
            for (int off = 1; off < 16; off <<= 1) {
                int nb = __shfl_up(inc, off, 16);
                if (l15 >= off) inc += nb;
            }

            if (hit) {
                const int idx = base[v] + inc - 1;          // exclusive slot
                const int gi  = mol * N_ATOMS + im;
                const int gj  = mol * N_ATOMS + jrow;
                const int pf  = inv_real_atoms[gi];
                const int ps  = inv_real_atoms[gj];
                const float* c1 = coords + 3 * (size_t)real_atoms[pf];
                const float* c2 = coords + 3 * (size_t)real_atoms[ps];
                const float dx = c2[0] - c1[0];
                const float dy = c2[1] - c1[1];
                const float dz = c2[2] - c1[2];
                dist[idx]               = sqrtf(dx * dx + dy * dy + dz * dz);
                pfirst[idx]             = pf;
                psecond[idx]            = ps;
                pcoord[3 * (size_t)idx + 0] = dx;
                pcoord[3 * (size_t)idx + 1] = dy;
                pcoord[3 * (size_t)idx + 2] = dz;
            }
            base[v] += __shfl(inc, 15, 16);                 // row total this tile
        }
    }
}

// ---------------------------------------------------------------------------
extern "C" void kernel_launch(void* const* d_in, const int* in_sizes, int n_in,
                              void* d_out, int out_size, void* d_ws, size_t ws_size,
                              hipStream_t stream)
{
    const float*         coords    = (const float*)d_in[0];
    const unsigned char* nonblank  = (const unsigned char*)d_in[1];
    const int*           real_at   = (const int*)d_in[2];
    const int*           inv_real  = (const int*)d_in[3];

    float* out     = (float*)d_out;
    float* dist    = out;
    int*   pfirst  = (int*)(out + MAX_PAIRS);
    int*   psecond = (int*)(out + 2 * MAX_PAIRS);
    float* pcoord  = out + 3 * MAX_PAIRS;

    int* counts  = (int*)d_ws;            // NROW ints
    int* offsets = counts + NROW;         // NROW ints (256KB total scratch)

    // 1) zero-fill output (store-bandwidth critical: ~403 MB of mostly padding)
    const long long n  = (long long)out_size;
    const long long n4 = n >> 2;
    zero_out<<<2048, 256, 0, stream>>>((float4*)d_out, n4, out, n);

    // 2) per-row neighbor counts via f32 WMMA Gram tiles
    dim3 grid(N_ATOMS / 16, N_MOL);       // (32 itiles, 64 mols), 1 wave each
    pair_count_wmma<<<grid, 32, 0, stream>>>(coords, nonblank, counts);

    // 3) exclusive scan over 32768 rows (row-major == jnp.nonzero order)
    scan_rows<<<1, 1024, 0, stream>>>(counts, offsets);

    // 4) ordered emission
    pair_emit_wmma<<<grid, 32, 0, stream>>>(coords, nonblank, real_at, inv_real,
                                            offsets, dist, pfirst, psecond, pcoord);
}